// ContinuousDiffusionProcess_17463337025693
// MI455X (gfx1250) — compile-verified
//
#include <hip/hip_runtime.h>
#include <math.h>

// VP-SDE Euler-Maruyama forward scan for MI455X (gfx1250).
// Memory-bound streaming kernel: 808 MB of traffic -> ~35us floor @ 23.3 TB/s.
// Data movement via the CDNA5 async-copy engine (global_load_async_to_lds_b128,
// ASYNCcnt-tracked, 8-deep LDS ring) + non-temporal hints so 800 MB of
// single-touch traffic doesn't thrash the 192 MB L2. No matrix ops: a WMMA
// (lower-triangular GEMM) reformulation moves the same bytes but 50x the FLOPs.

typedef float v4f __attribute__((ext_vector_type(4)));
typedef int   v4i __attribute__((ext_vector_type(4)));

#define S_STEPS 100
#define DEPTH   8     // async-to-LDS ring depth (power of two)
#define BLK     256   // 8 waves per block

#if defined(__HIP_DEVICE_COMPILE__)
  #if __has_builtin(__builtin_amdgcn_global_load_async_to_lds_b128) && \
      __has_builtin(__builtin_amdgcn_s_wait_asynccnt)
    #define HAVE_ASYNC 1
  #else
    #define HAVE_ASYNC 0
    #warning "gfx1250: async builtins NOT found, falling back to direct NT loads"
  #endif
#else
  #define HAVE_ASYNC 0
#endif

#if HAVE_ASYNC
// Exact parameter types per clang-22: (v4i AS1*, v4i AS3*, Ii offset, Ii cpol).
typedef __attribute__((address_space(1))) v4i gv4i;
typedef __attribute__((address_space(3))) v4i lv4i;

// Address-space casts via integer round-trip (generic LDS addr[31:0] == LDS offset).
__device__ __forceinline__ void async_ld16(const v4f* g, v4f* l) {
  __builtin_amdgcn_global_load_async_to_lds_b128(
      (gv4i*)(unsigned long long)g,
      (lv4i*)(unsigned int)(unsigned long long)l,
      /*offset=*/0, /*cpol=*/0);
}

// s_wait_asynccnt needs an ICE argument; used only in the (unrolled) epilogue,
// where the constant argument folds the switch away.
__device__ __forceinline__ void wait_async_le(int n) {
  switch (n) {
    case 0: __builtin_amdgcn_s_wait_asynccnt(0); break;
    case 1: __builtin_amdgcn_s_wait_asynccnt(1); break;
    case 2: __builtin_amdgcn_s_wait_asynccnt(2); break;
    case 3: __builtin_amdgcn_s_wait_asynccnt(3); break;
    case 4: __builtin_amdgcn_s_wait_asynccnt(4); break;
    case 5: __builtin_amdgcn_s_wait_asynccnt(5); break;
    case 6: __builtin_amdgcn_s_wait_asynccnt(6); break;
    default: __builtin_amdgcn_s_wait_asynccnt(7); break;
  }
}
#endif

// Uniform (wave-scalar) step coefficients; mirrors reference math.
__device__ __forceinline__ void step_coeffs(int t, float& a, float& b) {
  const float dt      = 1.0f / (float)S_STEPS;
  const float sqrt_dt = __builtin_sqrtf(dt);
  const float ntc     = (float)t / (float)S_STEPS;
  const float beta    = 0.1f + ntc * (20.0f - 0.1f);
  a = -0.5f * beta * dt;                  // drift*dt factor
  b = __builtin_sqrtf(beta) * sqrt_dt;    // diffusion factor
}

__global__ __launch_bounds__(BLK) void vpsde_scan_kernel(
    const float* __restrict__ xf, const float* __restrict__ nf,
    float* __restrict__ of, int NV)
{
  const int p = blockIdx.x * BLK + (int)threadIdx.x;
  if (p >= NV) return;
  const int tid = threadIdx.x;

  const v4f* __restrict__ x     = (const v4f*)xf;
  const v4f* __restrict__ noise = (const v4f*)nf;
  v4f* __restrict__ out         = (v4f*)of;

#if HAVE_ASYNC
  __shared__ v4f stage[DEPTH][BLK];   // 32 KB / block (WGP has 320 KB)
#endif

  // Slab 0 is the initial condition.
  v4f xv = x[p];
  __builtin_nontemporal_store(xv, &out[p]);

#if HAVE_ASYNC
  #pragma unroll
  for (int i = 0; i < DEPTH; ++i)
    async_ld16(&noise[(size_t)i * NV + p], &stage[i][tid]);

  // Main loop: steady state, constant wait threshold (DEPTH in flight).
  for (int t = 0; t < S_STEPS - DEPTH; ++t) {
    float a, b; step_coeffs(t, a, b);
    __builtin_amdgcn_s_wait_asynccnt(DEPTH - 1);  // oldest copy (stage t%D) done
    __asm__ volatile("" ::: "memory");            // keep LDS read below the wait
    v4f nz = stage[t & (DEPTH - 1)][tid];
    __asm__ volatile("" : "+v"(nz));              // materialize before slot reuse
    async_ld16(&noise[(size_t)(t + DEPTH) * NV + p], &stage[t & (DEPTH - 1)][tid]);
    xv = (xv + a * xv) + b * nz;
    __builtin_nontemporal_store(xv, &out[(size_t)(t + 1) * NV + p]);
  }

  // Epilogue: drain the ring, decreasing wait thresholds (fold to constants).
  #pragma unroll
  for (int k = 0; k < DEPTH; ++k) {
    const int t = S_STEPS - DEPTH + k;
    float a, b; step_coeffs(t, a, b);
    wait_async_le(DEPTH - 1 - k);
    __asm__ volatile("" ::: "memory");
    v4f nz = stage[t & (DEPTH - 1)][tid];
    xv = (xv + a * xv) + b * nz;
    __builtin_nontemporal_store(xv, &out[(size_t)(t + 1) * NV + p]);
  }
#else
  for (int t = 0; t < S_STEPS; ++t) {
    float a, b; step_coeffs(t, a, b);
    v4f nz = __builtin_nontemporal_load(&noise[(size_t)t * NV + p]);
    xv = (xv + a * xv) + b * nz;
    __builtin_nontemporal_store(xv, &out[(size_t)(t + 1) * NV + p]);
  }
#endif
}

extern "C" void kernel_launch(void* const* d_in, const int* in_sizes, int n_in,
                              void* d_out, int out_size, void* d_ws, size_t ws_size,
                              hipStream_t stream) {
  (void)n_in; (void)out_size; (void)d_ws; (void)ws_size;
  const float* x     = (const float*)d_in[0];
  const float* noise = (const float*)d_in[1];
  float* out         = (float*)d_out;

  const int N  = in_sizes[0];        // 64*256*64 = 1048576 floats
  const int NV = N / 4;              // float4 columns
  const int blocks = (NV + BLK - 1) / BLK;

  vpsde_scan_kernel<<<blocks, BLK, 0, stream>>>(x, noise, out, NV);
}